// Plastic_SimpleModulated_RNN_60163901882591
// MI455X (gfx1250) — compile-verified
//
#include <hip/hip_runtime.h>
#include <hip/hip_bf16.h>

#define B_   128
#define IN_  256
#define OUT_ 256
#define HS_  512

typedef __attribute__((ext_vector_type(2))) float v2f;
typedef __attribute__((ext_vector_type(4))) float v4f;
typedef __attribute__((ext_vector_type(8))) float v8f;

__device__ __forceinline__ float wave_sum(float v) {
    v += __shfl_xor(v, 16, 32);
    v += __shfl_xor(v, 8, 32);
    v += __shfl_xor(v, 4, 32);
    v += __shfl_xor(v, 2, 32);
    v += __shfl_xor(v, 1, 32);
    return v;
}

// ---------------------------------------------------------------------------
// Kernel 1: acc[b,i] = (inputs @ i2h_w.T)[b,i] + i2h_b[i] + (hidden @ w.T)[b,i]
// One wave per 16x16 output tile, V_WMMA_F32_16X16X4_F32, K-step = 4.
// A-operand lane layout: M = lane%16, K = 2*(lane/16) + {0,1}  -> v2f
// B-operand lane layout: N = lane%16, K = 2*(lane/16) + {0,1}  -> v2f
// D layout: vgpr t, lane l -> row M = t + 8*(l/16), col N = l%16
// ---------------------------------------------------------------------------
__global__ __launch_bounds__(32)
void k_pre_rec1(const float* __restrict__ inputs, const float* __restrict__ i2h_w,
                const float* __restrict__ i2h_b, const float* __restrict__ hidden,
                const float* __restrict__ w, float* __restrict__ acc) {
    const int n0 = blockIdx.x * 16;
    const int m0 = blockIdx.y * 16;
    const int lane = threadIdx.x;
    const int r = lane & 15;
    const int h = lane >> 4;

    v8f c = {};
    {
        const float* ap = inputs + (m0 + r) * IN_ + 2 * h;
        const float* bp = i2h_w  + (n0 + r) * IN_ + 2 * h;
#pragma unroll 4
        for (int k = 0; k < IN_; k += 4) {
            v2f av = *(const v2f*)(ap + k);
            v2f bv = *(const v2f*)(bp + k);
            c = __builtin_amdgcn_wmma_f32_16x16x4_f32(false, av, false, bv,
                                                      (short)0, c, false, false);
        }
    }
    {
        const float* ap = hidden + (m0 + r) * HS_ + 2 * h;
        const float* bp = w      + (n0 + r) * HS_ + 2 * h;
#pragma unroll 4
        for (int k = 0; k < HS_; k += 4) {
            v2f av = *(const v2f*)(ap + k);
            v2f bv = *(const v2f*)(bp + k);
            c = __builtin_amdgcn_wmma_f32_16x16x4_f32(false, av, false, bv,
                                                      (short)0, c, false, false);
        }
    }
    const float bias = i2h_b[n0 + r];
#pragma unroll
    for (int t = 0; t < 8; ++t)
        acc[(m0 + t + 8 * h) * HS_ + n0 + r] = c[t] + bias;
}

// ---------------------------------------------------------------------------
// Kernel 2: hactiv[b,i] = tanh(acc[b,i] + sum_j alpha[i,j]*hebb[b,i,j]*hidden[b,j])
// One wave per (b,i) row; 128-bit loads; regular temporal hints so hebb
// (134 MB) lands in the 192 MB L2 for reuse by kernel 5.
// ---------------------------------------------------------------------------
__global__ __launch_bounds__(256)
void k_rec2_hactiv(const float* __restrict__ hebb, const float* __restrict__ alpha,
                   const float* __restrict__ hidden, const float* __restrict__ acc,
                   float* __restrict__ hactiv) {
    const int wave = (blockIdx.x * 256 + threadIdx.x) >> 5;   // == b*HS_ + i
    const int lane = threadIdx.x & 31;
    const int b = wave >> 9;
    const int i = wave & (HS_ - 1);

    const float* hrow = hebb + (size_t)wave * HS_;
    const float* arow = alpha + (size_t)i * HS_;
    const float* hid  = hidden + (size_t)b * HS_;

    float s = 0.f;
#pragma unroll
    for (int j = lane * 4; j < HS_; j += 128) {
        __builtin_prefetch(hrow + j + 128, 0, 0);
        v4f hb = *(const v4f*)(hrow + j);
        v4f al = *(const v4f*)(arow + j);
        v4f hd = *(const v4f*)(hid + j);
        s += al.x * hb.x * hd.x + al.y * hb.y * hd.y
           + al.z * hb.z * hd.z + al.w * hb.w * hd.w;
    }
    s = wave_sum(s);
    if (lane == 0)
        hactiv[wave] = tanhf(acc[wave] + s);
}

// ---------------------------------------------------------------------------
// Kernel 3: valueout[b] = hactiv[b,:].h2v_w + h2v_b ; da[b] = tanh(.h2da_w + b)
// One wave per batch sample.
// ---------------------------------------------------------------------------
__global__ __launch_bounds__(256)
void k_da_value(const float* __restrict__ hactiv, const float* __restrict__ h2v_w,
                const float* __restrict__ h2v_b, const float* __restrict__ h2da_w,
                const float* __restrict__ h2da_b, float* __restrict__ valueout,
                float* __restrict__ da) {
    const int b = (blockIdx.x * 256 + threadIdx.x) >> 5;
    const int lane = threadIdx.x & 31;
    float sv = 0.f, sd = 0.f;
#pragma unroll
    for (int j = lane; j < HS_; j += 32) {
        float ha = hactiv[b * HS_ + j];
        sv += ha * h2v_w[j];
        sd += ha * h2da_w[j];
    }
    sv = wave_sum(sv);
    sd = wave_sum(sd);
    if (lane == 0) {
        valueout[b] = sv + h2v_b[0];
        da[b] = tanhf(sd + h2da_b[0]);
    }
}

// ---------------------------------------------------------------------------
// Kernel 4: activout = hactiv @ h2o_w.T + h2o_b  (WMMA f32 16x16x4)
// ---------------------------------------------------------------------------
__global__ __launch_bounds__(32)
void k_activout(const float* __restrict__ hactiv, const float* __restrict__ h2o_w,
                const float* __restrict__ h2o_b, float* __restrict__ activout) {
    const int n0 = blockIdx.x * 16;
    const int m0 = blockIdx.y * 16;
    const int lane = threadIdx.x;
    const int r = lane & 15;
    const int h = lane >> 4;

    v8f c = {};
    const float* ap = hactiv + (m0 + r) * HS_ + 2 * h;
    const float* bp = h2o_w  + (n0 + r) * HS_ + 2 * h;
#pragma unroll 4
    for (int k = 0; k < HS_; k += 4) {
        v2f av = *(const v2f*)(ap + k);
        v2f bv = *(const v2f*)(bp + k);
        c = __builtin_amdgcn_wmma_f32_16x16x4_f32(false, av, false, bv,
                                                  (short)0, c, false, false);
    }
    const float bias = h2o_b[n0 + r];
#pragma unroll
    for (int t = 0; t < 8; ++t)
        activout[(m0 + t + 8 * h) * OUT_ + n0 + r] = c[t] + bias;
}

// ---------------------------------------------------------------------------
// Kernel 5: hebb_new = clip(hebb + da[b]*hactiv[b,i]*hidden[b,j], -1, 1)
// hebb re-read hits L2 (loaded by kernel 2); the 134 MB hebb_new stream is
// stored non-temporally so it does not evict hebb/other data from L2.
// ---------------------------------------------------------------------------
__global__ __launch_bounds__(256)
void k_hebb_update(const float* __restrict__ hebb, const float* __restrict__ hidden,
                   const float* __restrict__ hactiv, const float* __restrict__ da,
                   float* __restrict__ hebb_new) {
    const size_t t = (size_t)blockIdx.x * 256 + threadIdx.x;  // float4 index
    const int j4 = (int)(t & 127);
    const int i  = (int)((t >> 7) & (HS_ - 1));
    const int b  = (int)(t >> 16);

    const float scale = da[b] * hactiv[b * HS_ + i];
    v4f hd = *(const v4f*)(hidden + b * HS_ + j4 * 4);
    v4f hb = *(const v4f*)(hebb + t * 4);
    v4f o;
    o.x = fminf(1.f, fmaxf(-1.f, hb.x + scale * hd.x));
    o.y = fminf(1.f, fmaxf(-1.f, hb.y + scale * hd.y));
    o.z = fminf(1.f, fmaxf(-1.f, hb.z + scale * hd.z));
    o.w = fminf(1.f, fmaxf(-1.f, hb.w + scale * hd.w));
    __builtin_nontemporal_store(o, (v4f*)(hebb_new + t * 4));
}

extern "C" void kernel_launch(void* const* d_in, const int* in_sizes, int n_in,
                              void* d_out, int out_size, void* d_ws, size_t ws_size,
                              hipStream_t stream) {
    const float* inputs = (const float*)d_in[0];
    const float* hidden = (const float*)d_in[1];
    const float* hebb   = (const float*)d_in[2];
    const float* i2h_w  = (const float*)d_in[3];
    const float* i2h_b  = (const float*)d_in[4];
    const float* w      = (const float*)d_in[5];
    const float* alpha  = (const float*)d_in[6];
    const float* h2o_w  = (const float*)d_in[7];
    const float* h2o_b  = (const float*)d_in[8];
    const float* h2v_w  = (const float*)d_in[9];
    const float* h2v_b  = (const float*)d_in[10];
    const float* h2da_w = (const float*)d_in[11];
    const float* h2da_b = (const float*)d_in[12];

    float* out      = (float*)d_out;
    float* activout = out;                       // B*OUT           = 32768
    float* valueout = out + B_ * OUT_;           // B               = 128
    float* hactiv   = valueout + B_;             // B*HS            = 65536
    float* hebb_new = hactiv + B_ * HS_;         // B*HS*HS

    float* acc = (float*)d_ws;                   // B*HS floats
    float* da  = acc + B_ * HS_;                 // B floats

    k_pre_rec1<<<dim3(HS_ / 16, B_ / 16), 32, 0, stream>>>(inputs, i2h_w, i2h_b,
                                                           hidden, w, acc);
    k_rec2_hactiv<<<dim3((B_ * HS_) / 8), 256, 0, stream>>>(hebb, alpha, hidden,
                                                            acc, hactiv);
    k_da_value<<<dim3(B_ / 8), 256, 0, stream>>>(hactiv, h2v_w, h2v_b,
                                                 h2da_w, h2da_b, valueout, da);
    k_activout<<<dim3(OUT_ / 16, B_ / 16), 32, 0, stream>>>(hactiv, h2o_w,
                                                            h2o_b, activout);
    k_hebb_update<<<dim3((B_ * HS_ * HS_ / 4) / 256), 256, 0, stream>>>(
        hebb, hidden, hactiv, da, hebb_new);
}